// LinkPredictor_893353198524
// MI455X (gfx1250) — compile-verified
//
#include <hip/hip_runtime.h>

// LinkPredictor edge scorer for MI455X (gfx1250, wave32, WMMA).
// score[e] = W2 . relu(W1 @ concat(h[src[e]], h[dst[e]]) + b1) + b2
//
//  - f32 WMMA 16x16x4 (matrix pipe, full fp32 numerics, RNE).
//  - A = W1 slice (16 hidden rows/wave) register-stationary: 64 v2f frags.
//  - B = X^T tile (16 edges) in LDS, 260-float padded rows -> all 64 banks
//    hit exactly once per ds_load_b64 fragment fetch.
//  - Double-buffered X tile + parity-buffered partials -> 1 barrier/tile,
//    next tile's L2 gather overlapped with the 64-WMMA chain.
//  - Two accumulator chains (even/odd k) to break WMMA D->C serialization.

typedef __attribute__((ext_vector_type(2))) float v2f;
typedef __attribute__((ext_vector_type(8))) float v8f;

#define DFEAT   128
#define KDIM    256            // 2*DFEAT
#define WAVES   16             // 16 waves x 16 hidden rows = HIDDEN 256
#define TPB     (WAVES * 32)
#define XSTRIDE 260            // padded X row stride (floats)

__global__ __launch_bounds__(TPB, 1)
void linkpred_wmma_kernel(const float* __restrict__ h,
                          const int*   __restrict__ src,
                          const int*   __restrict__ dst,
                          const float* __restrict__ W1w,   // [256][256] row-major
                          const float* __restrict__ W1b,   // [256]
                          const float* __restrict__ W2w,   // [256]
                          const float* __restrict__ W2b,   // [1]
                          float*       __restrict__ out,   // [E]
                          int nEdges)
{
    __shared__ float xs[2][16 * XSTRIDE];       // double-buffered X tiles
    __shared__ float wavePart[2][WAVES * 16];   // parity-buffered partials

    const int t     = threadIdx.x;
    const int w     = t >> 5;                   // wave id -> hidden slice [16w, 16w+16)
    const int l     = t & 31;                   // lane
    const int nh    = l & 15;                   // M (A) / N (B) lane index
    const int khalf = (l >> 4) * 2;             // K sub-offset: 0 or 2

    // ---- A-fragments: this wave's 16x256 slice of W1, held in VGPRs ----
    v2f aw[64];
    {
        const float* wrow = W1w + (size_t)(w * 16 + nh) * KDIM + khalf;
#pragma unroll
        for (int kk = 0; kk < 64; ++kk)
            aw[kk] = *(const v2f*)(wrow + kk * 4);
    }

    // ---- per-lane layer-2 constants for rows m = 16w + v + 8*(l>=16) ----
    float b1v[8], w2v[8];
#pragma unroll
    for (int v = 0; v < 8; ++v) {
        int m  = w * 16 + v + (khalf << 2);
        b1v[v] = W1b[m];
        w2v[v] = W2w[m];
    }
    const float b2 = W2b[0];

    const int nTiles = (nEdges + 15) >> 4;
    const int stride = gridDim.x;
    int tile = blockIdx.x;
    int buf  = 0;

    // ---- prologue: gather first tile into buffer 0 ----
    if (tile < nTiles) {
        int e = tile * 16 + w;
        if (e >= nEdges) e = nEdges - 1;
        const int node  = (l < 16) ? src[e] : dst[e];
        const float4* p = (const float4*)(h + (size_t)node * DFEAT + nh * 8);
        float4 f0 = p[0];
        float4 f1 = p[1];
        float4* q = (float4*)&xs[0][w * XSTRIDE + l * 8];
        q[0] = f0;
        q[1] = f1;
    }
    __syncthreads();

    for (; tile < nTiles; tile += stride, buf ^= 1) {
        // ---- 1. issue next tile's gather (loads fly under the WMMA chain) ----
        const int  next     = tile + stride;
        const bool haveNext = next < nTiles;
        float4 g0, g1;
        if (haveNext) {
            int e = next * 16 + w;
            if (e >= nEdges) e = nEdges - 1;
            const int node  = (l < 16) ? src[e] : dst[e];
            const float4* p = (const float4*)(h + (size_t)node * DFEAT + nh * 8);
            g0 = p[0];
            g1 = p[1];
        }

        // ---- 2. layer 1: 64x WMMA f32 16x16x4, two independent acc chains ----
        v8f acc0 = {}, acc1 = {};
        const v2f* xrow = (const v2f*)&xs[buf][nh * XSTRIDE + khalf];
#pragma unroll
        for (int kk = 0; kk < 64; kk += 2) {
            v2f bA = xrow[kk * 2];              // float offset 4*kk
            v2f bB = xrow[kk * 2 + 2];          // float offset 4*kk+4
            acc0 = __builtin_amdgcn_wmma_f32_16x16x4_f32(
                       false, aw[kk],     false, bA, (short)0, acc0, false, false);
            acc1 = __builtin_amdgcn_wmma_f32_16x16x4_f32(
                       false, aw[kk + 1], false, bB, (short)0, acc1, false, false);
        }

        // ---- 3. stage next tile into the other LDS buffer ----
        if (haveNext) {
            float4* q = (float4*)&xs[buf ^ 1][w * XSTRIDE + l * 8];
            q[0] = g0;
            q[1] = g1;
        }

        // ---- 4. layer 2: relu + dot with this wave's w2 slice ----
        float s = 0.f;
#pragma unroll
        for (int v = 0; v < 8; ++v) {
            float hv = acc0[v] + acc1[v] + b1v[v];
            hv = hv > 0.f ? hv : 0.f;
            s += hv * w2v[v];
        }
        s += __shfl_xor(s, 16, 32);             // fold lane halves
        if (l < 16) wavePart[buf][w * 16 + nh] = s;

        __syncthreads();                        // X handoff + partial handoff

        // ---- 5. deterministic cross-wave reduce + store (wave 0) ----
        if (t < 16) {
            float sum = b2;
#pragma unroll
            for (int ww = 0; ww < WAVES; ++ww) sum += wavePart[buf][ww * 16 + t];
            const int e = tile * 16 + t;
            if (e < nEdges) out[e] = sum;
        }
    }
}

extern "C" void kernel_launch(void* const* d_in, const int* in_sizes, int n_in,
                              void* d_out, int out_size, void* d_ws, size_t ws_size,
                              hipStream_t stream) {
    const float* h   = (const float*)d_in[0];
    const int*   src = (const int*)  d_in[1];
    const int*   dst = (const int*)  d_in[2];
    const float* W1w = (const float*)d_in[3];
    const float* W1b = (const float*)d_in[4];
    const float* W2w = (const float*)d_in[5];
    const float* W2b = (const float*)d_in[6];
    float* out = (float*)d_out;

    const int nEdges = in_sizes[1];
    const int nTiles = (nEdges + 15) / 16;
    int grid = nTiles < 2048 ? nTiles : 2048;
    if (grid < 1) grid = 1;

    hipLaunchKernelGGL(linkpred_wmma_kernel, dim3(grid), dim3(TPB), 0, stream,
                       h, src, dst, W1w, W1b, W2w, W2b, out, nEdges);
}